// Fape_pairLoss_12128987644147
// MI455X (gfx1250) — compile-verified
//
#include <hip/hip_runtime.h>
#include <math.h>

// ---------------- constants from the reference ----------------
#define EP      1.0e-4f
#define EPMAX   30.0f
#define RTOL    1.0e-5f
#define ATOL    1.0e-8f
#define MAXP    16384          // pair capacity (actual band matrix gives 8156)

typedef __attribute__((ext_vector_type(2))) float v2f;
typedef __attribute__((ext_vector_type(8))) float v8f;

// ---------------- small device helpers ----------------
__device__ __forceinline__ float dot3(const float* a, const float* b) {
    return a[0]*b[0] + a[1]*b[1] + a[2]*b[2];
}
__device__ __forceinline__ void cross3(const float* a, const float* b, float* c) {
    c[0] = a[1]*b[2] - a[2]*b[1];
    c[1] = a[2]*b[0] - a[0]*b[2];
    c[2] = a[0]*b[1] - a[1]*b[0];
}

// _rigid_from_3 for one residue: x = 3 points (9 floats), R rows = e1,e2,e3, t = x2
__device__ __forceinline__ void rigid_from_3(const float* x, float* R, float* t) {
    const float* x1 = x;
    const float* x2 = x + 3;
    const float* x3 = x + 6;
    float v1[3] = {x3[0]-x2[0], x3[1]-x2[1], x3[2]-x2[2]};
    float v2[3] = {x1[0]-x2[0], x1[1]-x2[1], x1[2]-x2[2]};
    float n1 = sqrtf(dot3(v1, v1)) + 0.001f;
    float e1[3] = {v1[0]/n1, v1[1]/n1, v1[2]/n1};
    float pr = dot3(e1, v2);
    float u2[3] = {v2[0]-e1[0]*pr, v2[1]-e1[1]*pr, v2[2]-e1[2]*pr};
    float n2 = sqrtf(dot3(u2, u2)) + 1e-8f;
    float e2[3] = {u2[0]/n2, u2[1]/n2, u2[2]/n2};
    float e3[3];
    cross3(e1, e2, e3);
    R[0]=e1[0]; R[1]=e1[1]; R[2]=e1[2];
    R[3]=e2[0]; R[4]=e2[1]; R[5]=e2[2];
    R[6]=e3[0]; R[7]=e3[1]; R[8]=e3[2];
    t[0]=x2[0]; t[1]=x2[1]; t[2]=x2[2];
}

// _merge_frames for one (A,B) pair
__device__ __forceinline__ void merge_frames(const float* RA, const float* OA,
                                             const float* RB, const float* OB,
                                             float* R, float* O) {
    bool close = true;
    #pragma unroll
    for (int k = 0; k < 3; ++k)
        close = close && (fabsf(OA[k]-OB[k]) <= ATOL + RTOL*fabsf(OB[k]));

    float d[3] = {OB[0]-OA[0], OB[1]-OA[1], OB[2]-OA[2]};
    float nd = sqrtf(dot3(d, d));
    float snd = (nd == 0.f) ? 1.f : nd;
    float X[3] = {d[0]/snd, d[1]/snd, d[2]/snd};

    float Za[3] = {RA[6]+RB[6], RA[7]+RB[7], RA[8]+RB[8]};
    float Zs[3] = {RA[6]-RB[6], RA[7]-RB[7], RA[8]-RB[8]};
    float na = sqrtf(dot3(Za, Za));
    float ns = sqrtf(dot3(Zs, Zs));
    float sna = (na == 0.f) ? 1.f : na;
    float sns = (ns == 0.f) ? 1.f : ns;
    float Zn[3];
    if (na > ns) { Zn[0]=Za[0]/sna; Zn[1]=Za[1]/sna; Zn[2]=Za[2]/sna; }
    else         { Zn[0]=Zs[0]/sns; Zn[1]=Zs[1]/sns; Zn[2]=Zs[2]/sns; }

    float Y[3];
    cross3(Zn, X, Y);
    bool nan_bad = (Y[0] != Y[0]) || (Y[1] != Y[1]) || (Y[2] != Y[2]);
    float cxy[3];
    cross3(X, Y, cxy);
    if (dot3(cxy, Zn) < 0.f) { Y[0] = -Y[0]; Y[1] = -Y[1]; Y[2] = -Y[2]; }

    bool bad = close || (nd == 0.f) || (na == 0.f) || (ns == 0.f) || nan_bad;
    if (bad) {
        #pragma unroll
        for (int k = 0; k < 9; ++k) R[k] = RA[k];
        O[0]=OA[0]; O[1]=OA[1]; O[2]=OA[2];
    } else {
        R[0]=X[0];  R[1]=X[1];  R[2]=X[2];
        R[3]=Y[0];  R[4]=Y[1];  R[5]=Y[2];
        R[6]=Zn[0]; R[7]=Zn[1]; R[8]=Zn[2];
        O[0]=0.5f*(OA[0]+OB[0]); O[1]=0.5f*(OA[1]+OB[1]); O[2]=0.5f*(OA[2]+OB[2]);
    }
}

// Frame record (32 floats): M[3][8] (combined [R | -S], K-padded), c[3], valid, pad[4]
__device__ __forceinline__ void write_frame(float* fr, const float* R, const float* t,
                                            const float* S, const float* u) {
    #pragma unroll
    for (int e = 0; e < 3; ++e) {
        fr[e*8+0] =  R[e*3+0]; fr[e*8+1] =  R[e*3+1]; fr[e*8+2] =  R[e*3+2];
        fr[e*8+3] = -S[e*3+0]; fr[e*8+4] = -S[e*3+1]; fr[e*8+5] = -S[e*3+2];
        fr[e*8+6] = 0.f;       fr[e*8+7] = 0.f;
        fr[24+e]  = dot3(R + e*3, t) - dot3(S + e*3, u);
    }
    fr[27] = 1.f;
    fr[28] = 0.f; fr[29] = 0.f; fr[30] = 0.f; fr[31] = 0.f;
}

// ---------------- kernels ----------------

// Zero counter + frame array (determinism across graph replays: ws not re-poisoned by harness)
__global__ void fape_init(int* counter, float* frames, int nFrameFloats) {
    int tid = blockIdx.x * blockDim.x + threadIdx.x;
    if (tid == 0) *counter = 0;
    for (int k = tid; k < nFrameFloats; k += gridDim.x * blockDim.x) frames[k] = 0.f;
}

// Compact (i,j) pairs with triu(matrix,1)==1 (order-free: final mean is order-invariant)
__global__ void fape_pairs(const int* __restrict__ matrix, int L,
                           int* counter, int* pairs) {
    int tid = blockIdx.x * blockDim.x + threadIdx.x;
    if (tid >= L * L) return;
    int i = tid / L;
    int j = tid - i * L;
    if (j > i && matrix[tid] == 1) {
        int idx = atomicAdd(counter, 1);
        if (idx < MAXP) { pairs[2*idx] = i; pairs[2*idx+1] = j; }
    }
}

// Per-residue: target rigid frames + base merged-frame records + point matrix Z
__global__ void fape_base(const float* __restrict__ coor, const float* __restrict__ rot,
                          const float* __restrict__ trans, const float* __restrict__ target,
                          int L, float* t_rot0, float* t_tran0, float* Z, float* frames) {
    int i = blockIdx.x * blockDim.x + threadIdx.x;
    if (i >= L) return;
    float R[9], t[3];
    rigid_from_3(target + (size_t)i*9, R, t);
    #pragma unroll
    for (int k = 0; k < 9; ++k) t_rot0[(size_t)i*9 + k] = R[k];
    #pragma unroll
    for (int k = 0; k < 3; ++k) t_tran0[(size_t)i*3 + k] = t[k];

    // base frame i: pred = (rot[i], trans[i]), target = (R, t)
    write_frame(frames + (size_t)i*32, rot + (size_t)i*9, trans + (size_t)i*3, R, t);

    // Z rows 3i..3i+2: [coor point | target point | 0 0]
    #pragma unroll
    for (int r = 0; r < 3; ++r) {
        float* zr = Z + (size_t)(3*i + r)*8;
        zr[0] = coor[(size_t)i*9 + r*3 + 0];
        zr[1] = coor[(size_t)i*9 + r*3 + 1];
        zr[2] = coor[(size_t)i*9 + r*3 + 2];
        zr[3] = target[(size_t)i*9 + r*3 + 0];
        zr[4] = target[(size_t)i*9 + r*3 + 1];
        zr[5] = target[(size_t)i*9 + r*3 + 2];
        zr[6] = 0.f; zr[7] = 0.f;
    }
}

// Merged pair frames for both pred and target rigs
__global__ void fape_pairframes(const float* __restrict__ rot, const float* __restrict__ trans,
                                const float* __restrict__ t_rot0, const float* __restrict__ t_tran0,
                                const int* __restrict__ pairs, const int* __restrict__ counter,
                                int L, float* frames) {
    int p = blockIdx.x * blockDim.x + threadIdx.x;
    int P = *counter; if (P > MAXP) P = MAXP;
    if (p >= P) return;
    int i = pairs[2*p], j = pairs[2*p+1];
    float Rp[9], Op[3], Rt[9], Ot[3];
    merge_frames(rot + (size_t)i*9, trans + (size_t)i*3,
                 rot + (size_t)j*9, trans + (size_t)j*3, Rp, Op);
    merge_frames(t_rot0 + (size_t)i*9, t_tran0 + (size_t)i*3,
                 t_rot0 + (size_t)j*9, t_tran0 + (size_t)j*3, Rt, Ot);
    write_frame(frames + (size_t)(L + p)*32, Rp, Op, Rt, Ot);
}

// Main: 16 points x 16 frames per wave; D_e = Z(16x8) * M_e(8x16) via 2x v_wmma_f32_16x16x4_f32
__global__ void fape_main(const float* __restrict__ Z, const float* __restrict__ frames,
                          float* __restrict__ partials, int NPT, int FT) {
    int wid  = (blockIdx.x * blockDim.x + threadIdx.x) >> 5;
    int lane = threadIdx.x & 31;
    int nWaves = NPT * FT;
    if (wid >= nWaves) return;               // uniform per wave
    int pt = wid % NPT;
    int ft = wid / NPT;

    int fcol = ft * 16 + (lane & 15);
    const float* fr = frames + (size_t)fcol * 32;
    float vflag = fr[27];

    float sum = 0.f;
    if (__ballot(vflag != 0.f) != 0ull) {
        int khalf = lane >> 4;               // wmma f32 K split across lane halves
        // A operand: point row (16x4 tiles, K = 2*khalf + v [+ 4 for step 1])
        int m = pt * 16 + (lane & 15);
        const float* zr = Z + (size_t)m * 8;
        v2f a0, a1;
        a0.x = zr[2*khalf + 0];     a0.y = zr[2*khalf + 1];
        a1.x = zr[4 + 2*khalf + 0]; a1.y = zr[4 + 2*khalf + 1];

        v8f acc[3];
        #pragma unroll
        for (int e = 0; e < 3; ++e) {
            v2f b0, b1;                      // B operand: col = frame, same K mapping
            b0.x = fr[e*8 + 2*khalf + 0];     b0.y = fr[e*8 + 2*khalf + 1];
            b1.x = fr[e*8 + 4 + 2*khalf + 0]; b1.y = fr[e*8 + 4 + 2*khalf + 1];
            v8f c = {};
            c = __builtin_amdgcn_wmma_f32_16x16x4_f32(false, a0, false, b0,
                                                      (short)0, c, false, false);
            c = __builtin_amdgcn_wmma_f32_16x16x4_f32(false, a1, false, b1,
                                                      (short)0, c, false, false);
            acc[e] = c;
        }

        float c0 = fr[24], c1 = fr[25], c2 = fr[26];
        #pragma unroll
        for (int v = 0; v < 8; ++v) {
            float d0 = acc[0][v] - c0;
            float d1 = acc[1][v] - c1;
            float d2 = acc[2][v] - c2;
            float err = sqrtf(d0*d0 + d1*d1 + d2*d2 + 3.f*EP);
            sum += vflag * fminf(err, EPMAX);
        }
    }
    // wave32 reduction
    #pragma unroll
    for (int off = 16; off > 0; off >>= 1) sum += __shfl_xor(sum, off, 32);
    if (lane == 0) partials[wid] = sum;
}

// Deterministic fixed-order reduction + mean
__global__ void fape_final(const float* __restrict__ partials, int n,
                           const int* __restrict__ counter, int L, int N, float* out) {
    __shared__ float sm[256];
    float s = 0.f;
    for (int i = threadIdx.x; i < n; i += blockDim.x) s += partials[i];
    sm[threadIdx.x] = s;
    __syncthreads();
    for (int stride = 128; stride > 0; stride >>= 1) {
        if ((int)threadIdx.x < stride) sm[threadIdx.x] += sm[threadIdx.x + stride];
        __syncthreads();
    }
    if (threadIdx.x == 0) {
        int P = *counter; if (P > MAXP) P = MAXP;
        double denom = (double)N * (double)(L + P);
        out[0] = (float)((double)sm[0] / denom);
    }
}

// ---------------- launch ----------------
extern "C" void kernel_launch(void* const* d_in, const int* in_sizes, int n_in,
                              void* d_out, int out_size, void* d_ws, size_t ws_size,
                              hipStream_t stream) {
    const float* coor   = (const float*)d_in[0];
    const float* rot    = (const float*)d_in[1];
    const float* trans  = (const float*)d_in[2];
    const float* target = (const float*)d_in[3];
    const int*   matrix = (const int*)d_in[4];

    int L = in_sizes[2] / 3;          // trans is (L,3)
    int N = 3 * L;                    // points
    int Fpad = L + MAXP;              // frame capacity (multiple of 16 for L=1024)
    int NPT = N / 16;
    int FT  = Fpad / 16;
    int nWaves = NPT * FT;

    // workspace layout
    int*   wsi      = (int*)d_ws;
    int*   counter  = wsi;                                 // [0]
    int*   pairs    = wsi + 16;                            // 2*MAXP ints
    float* wsf      = (float*)(wsi + 16 + 2*MAXP);
    float* t_rot0   = wsf;                                 // L*9
    float* t_tran0  = t_rot0 + (size_t)L*9;                // L*3
    float* Zb       = t_tran0 + (size_t)L*3;               // N*8
    float* frames   = Zb + (size_t)N*8;                    // Fpad*32
    float* partials = frames + (size_t)Fpad*32;            // nWaves

    int nFrameFloats = Fpad * 32;
    fape_init<<<(nFrameFloats + 255)/256, 256, 0, stream>>>(counter, frames, nFrameFloats);
    fape_pairs<<<(L*L + 255)/256, 256, 0, stream>>>(matrix, L, counter, pairs);
    fape_base<<<(L + 255)/256, 256, 0, stream>>>(coor, rot, trans, target, L,
                                                 t_rot0, t_tran0, Zb, frames);
    fape_pairframes<<<(MAXP + 255)/256, 256, 0, stream>>>(rot, trans, t_rot0, t_tran0,
                                                          pairs, counter, L, frames);
    fape_main<<<(nWaves*32 + 255)/256, 256, 0, stream>>>(Zb, frames, partials, NPT, FT);
    fape_final<<<1, 256, 0, stream>>>(partials, nWaves, counter, L, N, (float*)d_out);
}